// StandAloneSelfAttention_50835232916349
// MI455X (gfx1250) — compile-verified
//
#include <hip/hip_runtime.h>

// ---------------------------------------------------------------------------
// StandAloneSelfAttention for MI455X (gfx1250, wave32, WMMA)
//   x:[4,64,64,128]  Wq/Wk/Wv:[128,128]  b*:[128]  emb0:[64,7,1] emb1:[64,1,7]
//   out:[4,64,64,128] f32
// Projections use V_WMMA_F32_16X16X4_F32 (exact f32, K=4 per step, 32 steps).
// ---------------------------------------------------------------------------

typedef __attribute__((ext_vector_type(2))) float v2f;
typedef __attribute__((ext_vector_type(8))) float v8f;

#define NB    4
#define NH    64
#define NW    64
#define CDIM  128          // channels = filters
#define NPIX  (NB*NH*NW)   // 16384
#define HEADS 8
#define HSIZE 16
#define LDA   132          // padded LDS stride (4-bank skew per row -> conflict free)

// ---------------- QKV projection: D = X * W + b, tiled 16x16, K=128 --------
__global__ __launch_bounds__(256) void qkv_proj_kernel(
    const float* __restrict__ X,
    const float* __restrict__ Wq, const float* __restrict__ bq,
    const float* __restrict__ Wk, const float* __restrict__ bk,
    const float* __restrict__ Wv, const float* __restrict__ bv,
    float* __restrict__ Qo, float* __restrict__ Ko, float* __restrict__ Vo)
{
    __shared__ float As[16 * LDA];

    const int tid  = threadIdx.x;
    const int row0 = blockIdx.x * 16;

    const float* Wm; const float* bm; float* Om;
    if (blockIdx.y == 0)      { Wm = Wq; bm = bq; Om = Qo; }
    else if (blockIdx.y == 1) { Wm = Wk; bm = bk; Om = Ko; }
    else                      { Wm = Wv; bm = bv; Om = Vo; }

    // 16x128 A tile is contiguous in X -> fully coalesced cooperative load.
    #pragma unroll
    for (int idx = tid; idx < 16 * CDIM; idx += 256)
        As[(idx >> 7) * LDA + (idx & 127)] = X[row0 * CDIM + idx];
    __syncthreads();

    const int wave  = tid >> 5;
    const int lane  = tid & 31;
    const int m     = lane & 15;   // row (A/D) or col N (B/D)
    const int lhalf = lane >> 4;   // 0: lower K half / M=r ; 1: upper K half / M=r+8
    const int ncol0 = wave * 16;   // 8 waves cover N = 0..127

    v8f acc = {};
    #pragma unroll
    for (int kk = 0; kk < CDIM / 4; ++kk) {
        const int ka = kk * 4 + lhalf * 2;
        // A fragment (16x4 f32): lane m holds K=ka, ka+1 of row m
        v2f a;
        a.x = As[m * LDA + ka + 0];
        a.y = As[m * LDA + ka + 1];
        // B fragment (4x16 f32): lane holds N=ncol0+m, rows K=ka, ka+1
        v2f bfrag;
        bfrag.x = Wm[(ka + 0) * CDIM + ncol0 + m];
        bfrag.y = Wm[(ka + 1) * CDIM + ncol0 + m];
        acc = __builtin_amdgcn_wmma_f32_16x16x4_f32(
            /*neg_a=*/false, a, /*neg_b=*/false, bfrag,
            /*c_mod=*/(short)0, acc, /*reuse_a=*/false, /*reuse_b=*/false);
    }

    const float bias = bm[ncol0 + m];
    #pragma unroll
    for (int r = 0; r < 8; ++r) {
        const int row = row0 + r + lhalf * 8;          // D layout: VGPR r -> M=r / r+8
        Om[row * CDIM + ncol0 + m] = acc[r] + bias;
    }
}

// ---------------- per-pixel 7x7 local attention ---------------------------
__global__ __launch_bounds__(256) void attn_kernel(
    const float* __restrict__ Q, const float* __restrict__ Kb,
    const float* __restrict__ V,
    const float* __restrict__ emb0, const float* __restrict__ emb1,
    float* __restrict__ out)
{
    const int gid  = blockIdx.x * blockDim.x + threadIdx.x;  // 131072 threads
    const int head = gid & 7;
    const int pix  = gid >> 3;
    const int b    = pix >> 12;
    const int y    = (pix >> 6) & 63;
    const int x    = pix & 63;

    const int base = pix * CDIM + head * HSIZE;

    float q[HSIZE];
    #pragma unroll
    for (int d = 0; d < HSIZE; ++d) q[d] = Q[base + d];

    // emb: channel f = head*16+d; f<64 -> emb0[f*7+i]  else emb1[(f-64)*7+j]
    const bool lowhead = head < 4;
    const float* eptr = lowhead ? (emb0 + head * HSIZE * 7)
                                : (emb1 + (head - 4) * HSIZE * 7);

    float p[49];
    float vmax = -3.0e38f;
    #pragma unroll
    for (int i = 0; i < 7; ++i) {
        #pragma unroll
        for (int j = 0; j < 7; ++j) {
            const int yy = y + i - 3;
            const int xx = x + j - 3;
            const bool inb = ((unsigned)yy < 64u) && ((unsigned)xx < 64u);
            const float* kp = Kb + (((b << 12) + (yy << 6) + xx) * CDIM + head * HSIZE);
            const int eo = lowhead ? i : j;
            // NOTE: padded keys are zero BUT emb is still added -> OOB key = emb.
            float s = 0.f;
            #pragma unroll
            for (int d = 0; d < HSIZE; ++d) {
                const float kv = inb ? kp[d] : 0.f;
                s = fmaf(q[d], kv + eptr[d * 7 + eo], s);
            }
            p[i * 7 + j] = s;
            vmax = fmaxf(vmax, s);
        }
    }

    float sum = 0.f;
    #pragma unroll
    for (int t = 0; t < 49; ++t) {
        p[t] = __expf(p[t] - vmax);   // OOB positions DO count in the denominator
        sum += p[t];
    }

    float acc[HSIZE];
    #pragma unroll
    for (int d = 0; d < HSIZE; ++d) acc[d] = 0.f;
    #pragma unroll
    for (int i = 0; i < 7; ++i) {
        #pragma unroll
        for (int j = 0; j < 7; ++j) {
            const int yy = y + i - 3;
            const int xx = x + j - 3;
            if (((unsigned)yy < 64u) && ((unsigned)xx < 64u)) {  // OOB value = 0
                const float* vp = V + (((b << 12) + (yy << 6) + xx) * CDIM + head * HSIZE);
                const float w = p[i * 7 + j];
                #pragma unroll
                for (int d = 0; d < HSIZE; ++d) acc[d] = fmaf(w, vp[d], acc[d]);
            }
        }
    }

    const float inv = 1.0f / sum;
    #pragma unroll
    for (int d = 0; d < HSIZE; ++d) out[base + d] = acc[d] * inv;
}

// ---------------------------------------------------------------------------
extern "C" void kernel_launch(void* const* d_in, const int* in_sizes, int n_in,
                              void* d_out, int out_size, void* d_ws, size_t ws_size,
                              hipStream_t stream) {
    (void)in_sizes; (void)n_in; (void)out_size; (void)ws_size;

    const float* x    = (const float*)d_in[0];
    const float* Wq   = (const float*)d_in[1];
    const float* bq   = (const float*)d_in[2];
    const float* Wk   = (const float*)d_in[3];
    const float* bk   = (const float*)d_in[4];
    const float* Wv   = (const float*)d_in[5];
    const float* bv   = (const float*)d_in[6];
    const float* emb0 = (const float*)d_in[7];
    const float* emb1 = (const float*)d_in[8];

    float* ws = (float*)d_ws;
    float* Q = ws;                       // 16384*128 f32 = 8 MB
    float* K = ws + (size_t)NPIX * CDIM; // +8 MB
    float* V = ws + (size_t)2 * NPIX * CDIM;

    // 1024 M-tiles of 16 rows; y-dim selects Q/K/V matrix; 8 waves = 8 N-tiles
    qkv_proj_kernel<<<dim3(NPIX / 16, 3, 1), 256, 0, stream>>>(
        x, Wq, bq, Wk, bk, Wv, bv, Q, K, V);

    // one thread per (pixel, head)
    attn_kernel<<<dim3((NPIX * HEADS) / 256, 1, 1), 256, 0, stream>>>(
        Q, K, V, emb0, emb1, (float*)d_out);
}